// ImpGraphConvolution_9302899163451
// MI455X (gfx1250) — compile-verified
//
#include <hip/hip_runtime.h>

typedef __attribute__((ext_vector_type(2))) float v2f;
typedef __attribute__((ext_vector_type(4))) float v4f;
typedef __attribute__((ext_vector_type(8))) float v8f;

#define D 256
#define ROWS 32          // M rows per block = 2 WMMA M-tiles per wave
#define LDS_STRIDE 260   // 260 % 64 == 4 -> conflict-free b64 A-fetch

#define WMMA_F32(A, B, C) \
    __builtin_amdgcn_wmma_f32_16x16x4_f32(false, (A), false, (B), (short)0, (C), false, false)

// Fused dual-GEMM: h = x @ w_nbr (workspace), out = x @ w_own + bias (init d_out).
// x-tile staged in LDS once per block (x is streamed NT from HBM exactly once);
// each wave computes two 16x16 M-tiles for BOTH weights (4 accumulators),
// reusing the B fragments across M-tiles and the A fragments across weights.
__global__ __launch_bounds__(512) void ImpGC_gemm_dual(
    const float* __restrict__ x,
    const float* __restrict__ w_nbr,
    const float* __restrict__ w_own,
    const float* __restrict__ bias,
    float* __restrict__ h,
    float* __restrict__ out,
    int n_nodes)
{
    __shared__ float tile[ROWS * LDS_STRIDE];

    const int tid  = threadIdx.x;
    const int lane = tid & 31;
    const int wave = tid >> 5;            // 0..15 -> N tile
    const int m0   = blockIdx.x * ROWS;

    // ---- cooperative stage of x[m0 : m0+32, :] into LDS (non-temporal) ----
    {
        const int r  = tid >> 4;          // 0..31 row
        const int c  = (tid & 15) << 4;   // 0..240 step 16
        const int gr = m0 + r;
        if (gr < n_nodes) {
            const v4f* src = (const v4f*)(x + (size_t)gr * D + c);
            v4f* dst = (v4f*)(tile + r * LDS_STRIDE + c);
#pragma unroll
            for (int i = 0; i < 4; ++i)
                dst[i] = __builtin_nontemporal_load(src + i);
        }
    }
    __syncthreads();

    // A 16x4 f32 layout: lane l -> row M=l&15, K-pair 2*(l>>4)
    // B 4x16  f32 layout: lane l -> col N=l&15, K-pair 2*(l>>4)
    const int half = lane >> 4;
    const int lidx = lane & 15;
    const int koff = half * 2;
    const int n0   = wave * 16;
    const int ncol = n0 + lidx;

    const float* a0p = tile + lidx * LDS_STRIDE + koff;          // M-tile 0
    const float* a1p = tile + (16 + lidx) * LDS_STRIDE + koff;   // M-tile 1

    v8f acc_n0 = {}, acc_o0 = {}, acc_n1 = {}, acc_o1 = {};

#pragma unroll 4
    for (int k = 0; k < D; k += 4) {
        v2f a0 = *(const v2f*)(a0p + k);   // ds_load_b64, bank-conflict-free
        v2f a1 = *(const v2f*)(a1p + k);
        v2f bn, bo;
        bn.x = w_nbr[(size_t)(k + koff)     * D + ncol];
        bn.y = w_nbr[(size_t)(k + koff + 1) * D + ncol];
        bo.x = w_own[(size_t)(k + koff)     * D + ncol];
        bo.y = w_own[(size_t)(k + koff + 1) * D + ncol];
        acc_n0 = WMMA_F32(a0, bn, acc_n0);
        acc_n1 = WMMA_F32(a1, bn, acc_n1);
        acc_o0 = WMMA_F32(a0, bo, acc_o0);
        acc_o1 = WMMA_F32(a1, bo, acc_o1);
    }

    // C/D layout: VGPR r -> M = mbase + 8*half + r, N = n0 + lidx
    const float bv  = bias[ncol];
    const int mb0 = m0 + half * 8;
    const int mb1 = m0 + 16 + half * 8;
#pragma unroll
    for (int r = 0; r < 8; ++r) {
        if (mb0 + r < n_nodes) {
            size_t off = (size_t)(mb0 + r) * D + ncol;
            h[off]   = acc_n0[r];
            out[off] = acc_o0[r] + bv;
        }
        if (mb1 + r < n_nodes) {
            size_t off = (size_t)(mb1 + r) * D + ncol;
            h[off]   = acc_n1[r];
            out[off] = acc_o1[r] + bv;
        }
    }
}

// Edge scatter: out[dst] += adj * h[src].
// One wave per edge (wave-uniform edge metadata -> scalar loads),
// 8 channels per lane: 2x global_load_b128 + 8x global_atomic_add_f32.
// h (102 MB) and out (102 MB) stay largely L2-resident (192 MB L2).
__global__ __launch_bounds__(256) void ImpGC_scatter(
    const float* __restrict__ h,
    const int*   __restrict__ esrc,
    const int*   __restrict__ edst,
    const float* __restrict__ vals,
    float* __restrict__ out,
    int n_edges)
{
    const int lane  = threadIdx.x & 31;
    const int wv    = threadIdx.x >> 5;     // 0..7
    const int cbase = lane << 3;            // 8 floats per lane
    int e = blockIdx.x * 32 + wv;

#pragma unroll
    for (int it = 0; it < 4; ++it, e += 8) {
        if (e < n_edges) {
            const int   s = esrc[e];
            const int   d = edst[e];
            const float a = vals[e];
            const v4f h0 = *(const v4f*)(h + (size_t)s * D + cbase);
            const v4f h1 = *(const v4f*)(h + (size_t)s * D + cbase + 4);
            float* o = out + (size_t)d * D + cbase;
            unsafeAtomicAdd(o + 0, a * h0.x);
            unsafeAtomicAdd(o + 1, a * h0.y);
            unsafeAtomicAdd(o + 2, a * h0.z);
            unsafeAtomicAdd(o + 3, a * h0.w);
            unsafeAtomicAdd(o + 4, a * h1.x);
            unsafeAtomicAdd(o + 5, a * h1.y);
            unsafeAtomicAdd(o + 6, a * h1.z);
            unsafeAtomicAdd(o + 7, a * h1.w);
        }
    }
}

extern "C" void kernel_launch(void* const* d_in, const int* in_sizes, int n_in,
                              void* d_out, int out_size, void* d_ws, size_t ws_size,
                              hipStream_t stream)
{
    const float* x     = (const float*)d_in[0];
    const int*   esrc  = (const int*)  d_in[1];
    const int*   edst  = (const int*)  d_in[2];
    const float* vals  = (const float*)d_in[3];
    const float* w_nbr = (const float*)d_in[4];
    const float* w_own = (const float*)d_in[5];
    const float* bias  = (const float*)d_in[6];
    float* out = (float*)d_out;
    float* h   = (float*)d_ws;                  // n_nodes * 256 * 4 bytes

    const int n_nodes = in_sizes[0] / D;
    const int n_edges = in_sizes[1];

    const int mblocks = (n_nodes + ROWS - 1) / ROWS;
    ImpGC_gemm_dual<<<mblocks, 512, 0, stream>>>(x, w_nbr, w_own, bias, h, out, n_nodes);

    const int eblocks = (n_edges + 31) / 32;
    ImpGC_scatter<<<eblocks, 256, 0, stream>>>(h, esrc, edst, vals, out, n_edges);
}